// PairwiseRepresentation_4818953306747
// MI455X (gfx1250) — compile-verified
//
#include <hip/hip_runtime.h>
#include <cstdint>

// Problem constants from the reference: B=16, N=4096, K=128
#define B_  16
#define N_  4096
#define K_  128
#define ROWS_PER_BLOCK 64     // 8 row-groups x 8 iterations; 1024 blocks total
#define THREADS 256           // 8 wave32s; each wave covers one row (32 lanes x 4 k)
#define ROW_ITERS (ROWS_PER_BLOCK / 8)

typedef int   v4i __attribute__((ext_vector_type(4)));
typedef float v4f __attribute__((ext_vector_type(4)));

// Memory-bound gather kernel:
//  - positions[b] (48KB) staged into LDS once per block via async global->LDS DMA
//    (ASYNCcnt path), amortized over 64 rows -> 48 MiB total staging traffic
//  - one-touch streams (neighbors/mask/offsets/out) use TH=NT 128-bit accesses
//  - cell[b] is block-uniform -> scalar loads
__global__ __launch_bounds__(THREADS)
void PairwiseRepresentation_kernel(
    const float* __restrict__ positions,   // [B,N,3]
    const int*   __restrict__ neighbors,   // [B,N,K]
    const float* __restrict__ nmask,       // [B,N,K]
    const float* __restrict__ cell,        // [B,3,3]
    const float* __restrict__ offs,        // [B,N,K,3]
    float*       __restrict__ out)         // [B,N,K]
{
    __shared__ float spos[N_ * 3];         // 48 KB: positions for this block's batch

    const int t    = threadIdx.x;
    const int row0 = blockIdx.x * ROWS_PER_BLOCK;   // global row = b*N + n
    const int b    = row0 >> 12;                     // N=4096 -> uniform per block
    const int lane = t & 31;
    const int rgrp = t >> 5;                         // 0..7: row within iteration group

    // ---- stage positions[b] (49152 B) into LDS via async DMA ----
    // 256 threads x 12 iters x 16 B = 48 KB, ASYNCcnt-tracked.
    const float* posb = positions + (size_t)b * (N_ * 3);
    const uint32_t lds_base = (uint32_t)(uintptr_t)spos;  // low 32 bits = LDS byte addr
#pragma unroll
    for (int i = 0; i < 12; ++i) {
        uint32_t goff = (uint32_t)(t * 16 + i * 4096);
        uint32_t ldst = lds_base + goff;
        asm volatile("global_load_async_to_lds_b128 %0, %1, %2"
                     :: "v"(ldst), "v"(goff), "s"(posb)
                     : "memory");
    }

    // cell[b]: uniform address -> scalar loads, broadcast to lanes
    const float* cb = cell + b * 9;
    const float c00 = cb[0], c01 = cb[1], c02 = cb[2];
    const float c10 = cb[3], c11 = cb[4], c12 = cb[5];
    const float c20 = cb[6], c21 = cb[7], c22 = cb[8];

    // wait for this wave's async DMA, then make LDS visible workgroup-wide
    asm volatile("s_wait_asynccnt 0x0" ::: "memory");
    __syncthreads();

#pragma unroll 2
    for (int it = 0; it < ROW_ITERS; ++it) {
        const int  row = row0 + it * 8 + rgrp;
        const int  n   = row & (N_ - 1);
        const long idx = (long)row * K_ + lane * 4;

        // one-touch streams: 128-bit non-temporal loads
        const v4i nb = __builtin_nontemporal_load((const v4i*)(neighbors + idx));
        const v4f mk = __builtin_nontemporal_load((const v4f*)(nmask + idx));
        const v4f* op = (const v4f*)(offs + idx * 3);     // 48 B, 16 B aligned
        const v4f o0 = __builtin_nontemporal_load(op + 0);
        const v4f o1 = __builtin_nontemporal_load(op + 1);
        const v4f o2 = __builtin_nontemporal_load(op + 2);

        const float xi = spos[n * 3 + 0];
        const float yi = spos[n * 3 + 1];
        const float zi = spos[n * 3 + 2];

        // unpack [4][3] offsets from three float4s
        const int   jj[4] = { nb.x, nb.y, nb.z, nb.w };
        const float mm[4] = { mk.x, mk.y, mk.z, mk.w };
        const float ox[4] = { o0.x, o0.w, o1.z, o2.y };
        const float oy[4] = { o0.y, o1.x, o1.w, o2.z };
        const float oz[4] = { o0.z, o1.y, o2.x, o2.w };

        v4f res;
#pragma unroll
        for (int q = 0; q < 4; ++q) {
            const int j = jj[q];
            const float xj = spos[j * 3 + 0];
            const float yj = spos[j * 3 + 1];
            const float zj = spos[j * 3 + 2];
            // dist_vec = (pos_j - pos_i) + off @ cell  (out_e = sum_d off_d * cell[d][e])
            const float dx = xj - xi + ox[q] * c00 + oy[q] * c10 + oz[q] * c20;
            const float dy = yj - yi + ox[q] * c01 + oy[q] * c11 + oz[q] * c21;
            const float dz = zj - zi + ox[q] * c02 + oy[q] * c12 + oz[q] * c22;
            const float sq = dx * dx + dy * dy + dz * dz;
            res[q] = (mm[q] > 0.0f) ? __builtin_sqrtf(sq) : 0.0f;
        }
        __builtin_nontemporal_store(res, (v4f*)(out + idx));
    }
}

extern "C" void kernel_launch(void* const* d_in, const int* in_sizes, int n_in,
                              void* d_out, int out_size, void* d_ws, size_t ws_size,
                              hipStream_t stream) {
    (void)in_sizes; (void)n_in; (void)d_ws; (void)ws_size; (void)out_size;
    const float* positions = (const float*)d_in[0];
    const int*   neighbors = (const int*)  d_in[1];
    const float* nmask     = (const float*)d_in[2];
    const float* cell      = (const float*)d_in[3];
    const float* offs      = (const float*)d_in[4];
    float* out = (float*)d_out;

    const int total_rows = B_ * N_;                      // 65536
    dim3 grid(total_rows / ROWS_PER_BLOCK);              // 1024 blocks
    dim3 block(THREADS);                                 // 8 wave32s
    PairwiseRepresentation_kernel<<<grid, block, 0, stream>>>(
        positions, neighbors, nmask, cell, offs, out);
}